// MoE_32839319945482
// MI455X (gfx1250) — compile-verified
//
#include <hip/hip_runtime.h>
#include <math.h>

// MoE top-1 router + per-expert Linear(10,10) for MI455X (gfx1250, wave32).
//
// Bandwidth-bound: 4M tokens * (40B in + 40B out) = 336MB -> ~14.4us floor at
// 23.3 TB/s. Compute (~1 GFLOP fp32) is free, so we keep fp32 end-to-end and
// run the 10x10 matmul on the WMMA pipe (V_WMMA_F32_16X16X4_F32), which lets
// each wave keep both expert weight matrices stationary in 12 VGPRs.
//
// Exact identity used:  out = r0@W1 + r1@W2 + (b1+b2), where r0/r1 are the
// top-1 masked, gate-scaled token rows (masked rows contribute exact zeros
// to the accumulator).
//
// Round-1 fix: the tile-full test is wave-uniform, so the hot path now does
// 8 straight-line global_store_b32 with immediate offsets off one 64-bit base
// (no per-row exec juggling / 64-bit mads), and the prefetch is unconditional
// with a clamped tile index (speculative prefetch is safe).

typedef __attribute__((ext_vector_type(2))) float v2f;
typedef __attribute__((ext_vector_type(8))) float v8f;

#define WAVES_PER_BLOCK 8
#define THREADS (WAVES_PER_BLOCK * 32)

__global__ __launch_bounds__(THREADS) void moe_router_mlp_kernel(
    const float* __restrict__ x,
    const float* __restrict__ Wg,
    const float* __restrict__ W1,
    const float* __restrict__ b1,
    const float* __restrict__ W2,
    const float* __restrict__ b2,
    float* __restrict__ out,
    int n_tokens, int n_tiles)
{
    const int lane     = threadIdx.x & 31;
    const int waveBlk  = threadIdx.x >> 5;
    const int gwave    = blockIdx.x * WAVES_PER_BLOCK + waveBlk;
    const int wstride  = gridDim.x * WAVES_PER_BLOCK;

    const int n  = lane & 15;   // output column / B column for this lane
    const int kh = lane >> 4;   // 0 -> K pair {0,1}, 1 -> {2,3} within each K4 chunk

    // ---- per-wave setup (hoisted out of the streaming loop) ----
    // Gate weights: uniform addresses -> scalar loads into SGPRs.
    float wg0[10], wg1[10];
#pragma unroll
    for (int k = 0; k < 10; ++k) { wg0[k] = Wg[k * 2 + 0]; wg1[k] = Wg[k * 2 + 1]; }

    // Expert weight B-tiles, padded 12x16 (zeros for k>=10 or n>=10), as
    // three K=4 chunks. ISA B layout: lane l holds B[k0+2kh][n], B[k0+2kh+1][n].
    v2f B1[3], B2[3];
#pragma unroll
    for (int c = 0; c < 3; ++c) {
        const int k0 = 4 * c + 2 * kh;
        float e1x = 0.f, e1y = 0.f, e2x = 0.f, e2y = 0.f;
        if (n < 10) {
            if (k0 < 10)     { e1x = W1[k0 * 10 + n];       e2x = W2[k0 * 10 + n]; }
            if (k0 + 1 < 10) { e1y = W1[(k0 + 1) * 10 + n]; e2y = W2[(k0 + 1) * 10 + n]; }
        }
        B1[c].x = e1x; B1[c].y = e1y;
        B2[c].x = e2x; B2[c].y = e2y;
    }

    const float bias = (n < 10) ? (b1[n] + b2[n]) : 0.0f;

    // ---- stream 16-token tiles ----
    for (int t = gwave; t < n_tiles; t += wstride) {
        const int t0 = t * 16;
        const bool full = (t0 + 16 <= n_tokens);   // wave-uniform

        // Speculative prefetch of the next tile this wave will touch (640B),
        // clamped index so no branch / exec change is needed.
        int tn = t + wstride;
        if (tn > n_tiles - 1) tn = n_tiles - 1;    // uniform select, no divergence
        {
            const char* pf = (const char*)(x + (size_t)tn * 160);
            __builtin_prefetch(pf + (size_t)lane * 20, 0, 1);
        }

        // Each lane loads the full row of token m = lane&15 (lane pairs l and
        // l+16 duplicate the request; L1 absorbs it, HBM traffic unchanged).
        int rowi = t0 + n;
        if (!full && rowi > n_tokens - 1) rowi = n_tokens - 1;  // tail only
        const float* xr = x + (size_t)rowi * 10;

        float xv[12];
#pragma unroll
        for (int j = 0; j < 5; ++j) {
            v2f p = *reinterpret_cast<const v2f*>(xr + 2 * j);  // 8B-aligned b64 loads
            xv[2 * j] = p.x; xv[2 * j + 1] = p.y;
        }
        xv[10] = 0.f; xv[11] = 0.f;

        // Gate: scores, 2-way softmax, top-1 select (argmax ties -> expert 0).
        float s0 = 0.f, s1 = 0.f;
#pragma unroll
        for (int k = 0; k < 10; ++k) {
            s0 = fmaf(xv[k], wg0[k], s0);
            s1 = fmaf(xv[k], wg1[k], s1);
        }
        const float mx  = fmaxf(s0, s1);
        const float e0  = expf(s0 - mx);
        const float e1  = expf(s1 - mx);
        const float inv = 1.0f / (e0 + e1);
        const bool  top0 = (s0 >= s1);
        const float sel0 = top0 ? (e0 * inv) : 0.0f;
        const float sel1 = top0 ? 0.0f       : (e1 * inv);

        // C += A0@W1 + A1@W2 over three K=4 chunks.
        // ISA A layout: lane l holds A[m][k0+2kh], A[m][k0+2kh+1].
        v8f acc = {};
#pragma unroll
        for (int c = 0; c < 3; ++c) {
            const int k0 = 4 * c;
            const float ax = kh ? xv[k0 + 2] : xv[k0 + 0];
            const float ay = kh ? xv[k0 + 3] : xv[k0 + 1];
            v2f A0; A0.x = sel0 * ax; A0.y = sel0 * ay;
            v2f A1; A1.x = sel1 * ax; A1.y = sel1 * ay;
            acc = __builtin_amdgcn_wmma_f32_16x16x4_f32(
                false, A0, false, B1[c], (short)0, acc, false, false);
            acc = __builtin_amdgcn_wmma_f32_16x16x4_f32(
                false, A1, false, B2[c], (short)0, acc, false, false);
        }

        // Store: C VGPR r on lane l is element (M = 8*kh + r, N = n).
        if (full) {
            // Hot path: one exec mask (n<10), one 64-bit base, 8 stores with
            // immediate offsets (row stride 40B).
            if (n < 10) {
                float* op = out + (size_t)(t0 + 8 * kh) * 10 + n;
#pragma unroll
                for (int r = 0; r < 8; ++r)
                    op[r * 10] = acc[r] + bias;
            }
        } else {
            // Tail tile (never taken for N % 16 == 0): per-row bound check.
            if (n < 10) {
                const int rbase = t0 + 8 * kh;
#pragma unroll
                for (int r = 0; r < 8; ++r) {
                    const int row = rbase + r;
                    if (row < n_tokens)
                        out[(size_t)row * 10 + n] = acc[r] + bias;
                }
            }
        }
    }
}

extern "C" void kernel_launch(void* const* d_in, const int* in_sizes, int n_in,
                              void* d_out, int out_size, void* d_ws, size_t ws_size,
                              hipStream_t stream) {
    const float* x  = (const float*)d_in[0];
    const float* Wg = (const float*)d_in[1];
    const float* W1 = (const float*)d_in[2];
    const float* b1 = (const float*)d_in[3];
    const float* W2 = (const float*)d_in[4];
    const float* b2 = (const float*)d_in[5];
    float* out = (float*)d_out;

    const int n_tokens = in_sizes[0] / 10;
    const int n_tiles  = (n_tokens + 15) / 16;

    // ~8 tiles (128 tokens) per wave; cap grid so the loop amortizes setup.
    int blocks = (n_tiles + WAVES_PER_BLOCK * 8 - 1) / (WAVES_PER_BLOCK * 8);
    if (blocks > 4096) blocks = 4096;
    if (blocks < 1)    blocks = 1;

    moe_router_mlp_kernel<<<blocks, THREADS, 0, stream>>>(
        x, Wg, W1, b1, W2, b2, out, n_tokens, n_tiles);
}